// HomoGAT_6451040878628
// MI455X (gfx1250) — compile-verified
//
#include <hip/hip_runtime.h>
#include <hip/hip_bf16.h>
#include <math.h>

#define IN_DIM 519
#define HID 128
#define HEADS 4
#define NRXN 10000
#define NMET 20000
#define NNODE (NRXN + NMET)
#define NEDGE 100000
#define ETOT (2 * NEDGE + NNODE)
#define HC (HEADS * HID)   // 512

typedef float v2f __attribute__((ext_vector_type(2)));
typedef float v8f __attribute__((ext_vector_type(8)));

// ---------------------------------------------------------------- utilities
__device__ __forceinline__ float elu1(float x) { return x > 0.f ? x : expm1f(x); }

__device__ __forceinline__ float wave_sum(float v) {
  for (int off = 16; off; off >>= 1) v += __shfl_xor(v, off, 32);
  return v;
}

// order-preserving float <-> uint map for atomic segment-max
__device__ __forceinline__ unsigned fenc(float f) {
  unsigned b = __float_as_uint(f);
  return (b & 0x80000000u) ? ~b : (b | 0x80000000u);
}
__device__ __forceinline__ float fdec(unsigned u) {
  unsigned b = (u & 0x80000000u) ? (u ^ 0x80000000u) : ~u;
  return __uint_as_float(b);
}

__global__ void zero_f32(float* p, size_t n) {
  size_t i = (size_t)blockIdx.x * blockDim.x + threadIdx.x;
  if (i < n) p[i] = 0.f;
}

// ---------------------------------------------------------------- edge list
__global__ void build_edges(const int* __restrict__ rs, const int* __restrict__ rd,
                            const int* __restrict__ ms, const int* __restrict__ md,
                            int* __restrict__ srcA, int* __restrict__ dstA) {
  int i = blockIdx.x * blockDim.x + threadIdx.x;
  if (i < NEDGE) {
    srcA[i] = rs[i];
    dstA[i] = rd[i] + NRXN;
  } else if (i < 2 * NEDGE) {
    srcA[i] = ms[i - NEDGE] + NRXN;
    dstA[i] = md[i - NEDGE];
  } else if (i < ETOT) {
    int n = i - 2 * NEDGE;
    srcA[i] = n;
    dstA[i] = n;
  }
}

// ---------------------------------------------------------------- WMMA GEMMs
// fp32 WMMA 16x16x4. A-frag per-lane: lanes0-15 K=k0,k0+1; lanes16-31 K=k0+2,k0+3.
// B-frag: VGPR0 = row k0 (lanes0-15) / k0+2 (lanes16-31); VGPR1 = k0+1 / k0+3.
__device__ __forceinline__ v8f wmma4(v2f a, v2f b, v8f c) {
  return __builtin_amdgcn_wmma_f32_16x16x4_f32(false, a, false, b, (short)0, c, false, false);
}

// Projection GEMM: [N,519] @ [519,128] with structural zero-padding exploited.
// grid.x = NNODE/16 row tiles; 8 waves/block = 8 column tiles (128/16).
// Main K-loop is UNGUARDED (no divergent loads); odd tail handled once with
// clamped addresses + value selects (v_cndmask, no EXEC branching).
__global__ void proj_gemm_kernel(const float* __restrict__ xrxn,
                                 const float* __restrict__ xmet,
                                 const float* __restrict__ W,
                                 const float* __restrict__ bias,
                                 float* __restrict__ out) {
  const int tm = blockIdx.x;            // 0..1874
  const int tn = threadIdx.x >> 5;      // 0..7
  const int lane = threadIdx.x & 31;
  const int half = lane >> 4;
  const int l16 = lane & 15;
  const float* A;
  int lda, Keff;
  if (tm < NRXN / 16) { A = xrxn + (size_t)tm * 16 * 64; lda = 64; Keff = 64; }
  else                { A = xmet + (size_t)(tm * 16 - NRXN) * IN_DIM; lda = IN_DIM; Keff = IN_DIM; }
  const int col = tn * 16 + l16;
  const float* Arow = A + (size_t)l16 * lda;
  v8f acc = {};
  const int kmain = Keff & ~3;          // 64 or 516
#pragma unroll 4
  for (int k = 0; k < kmain; k += 4) {
    int ka = k + 2 * half;
    v2f av = {Arow[ka], Arow[ka + 1]};
    v2f bv = {W[(size_t)ka * HID + col], W[(size_t)(ka + 1) * HID + col]};
    acc = wmma4(av, bv, acc);
  }
  if (kmain < Keff) {                   // only for Keff=519: k = 516,517,518
    int ka = kmain + 2 * half;          // 516 or 518 (both valid rows)
    bool v1 = (ka + 1) < Keff;          // 517 valid, 519 not
    int ka1 = v1 ? (ka + 1) : ka;       // clamped address -> unconditional load
    float a0 = Arow[ka];
    float a1 = v1 ? Arow[ka1] : 0.f;
    float b0 = W[(size_t)ka * HID + col];
    float b1 = v1 ? W[(size_t)ka1 * HID + col] : 0.f;
    v2f av = {a0, a1};
    v2f bv = {b0, b1};
    acc = wmma4(av, bv, acc);
  }
  const float bb = bias[col];
  const int m0 = tm * 16;
  for (int v = 0; v < 8; ++v)
    out[(size_t)(m0 + v + 8 * half) * HID + col] = acc[v] + bb;
}

// Generic C[M,Nc] = A[M,K] @ B[K,Nc] + bias ; M%16==0, K%4==0, Nc%16==0.
// A row stride K is a multiple of 4 floats in all uses -> aligned v2f loads.
__global__ void gemm_wmma_bias(const float* __restrict__ A,
                               const float* __restrict__ B,
                               const float* __restrict__ bias,
                               float* __restrict__ C,
                               int M, int K, int Nc) {
  const int wave = blockIdx.x * (blockDim.x >> 5) + (threadIdx.x >> 5);
  const int lane = threadIdx.x & 31;
  const int ntiles = Nc >> 4;
  const int totalTiles = (M >> 4) * ntiles;
  if (wave >= totalTiles) return;                 // wave-uniform exit
  const int tm = wave / ntiles, tn = wave % ntiles;
  const int half = lane >> 4, l16 = lane & 15;
  const int m0 = tm << 4, n0 = tn << 4;
  const float* Arow = A + (size_t)(m0 + l16) * K + 2 * half;
  const float* Bcol = B + n0 + l16;
  v8f acc = {};
#pragma unroll 4
  for (int k = 0; k < K; k += 4) {
    int ka = k + 2 * half;
    v2f av = *(const v2f*)(Arow + k);             // global_load_b64
    v2f bv = {Bcol[(size_t)ka * Nc], Bcol[(size_t)(ka + 1) * Nc]};
    acc = wmma4(av, bv, acc);
  }
  const float bb = bias[n0 + l16];
  for (int v = 0; v < 8; ++v)
    C[(size_t)(m0 + v + 8 * half) * Nc + n0 + l16] = acc[v] + bb;
}

// ---------------------------------------------------------------- row LayerNorm (+ELU)
__global__ void ln_elu_rows(const float* __restrict__ in, const float* __restrict__ g,
                            const float* __restrict__ b, float* __restrict__ out, int nrows) {
  const int wave = blockIdx.x * (blockDim.x >> 5) + (threadIdx.x >> 5);
  const int lane = threadIdx.x & 31;
  if (wave >= nrows) return;
  const float* row = in + (size_t)wave * HID;
  float v[4], s = 0.f, s2 = 0.f;
  for (int j = 0; j < 4; ++j) { v[j] = row[lane + 32 * j]; s += v[j]; s2 += v[j] * v[j]; }
  s = wave_sum(s); s2 = wave_sum(s2);
  float mean = s * (1.f / HID);
  float var  = s2 * (1.f / HID) - mean * mean;
  float inv  = rsqrtf(var + 1e-5f);
  for (int j = 0; j < 4; ++j) {
    int c = lane + 32 * j;
    out[(size_t)wave * HID + c] = elu1((v[j] - mean) * inv * g[c] + b[c]);
  }
}

// ---------------------------------------------------------------- GATv2 edge ops
__global__ void edge_logits_kernel(const float* __restrict__ xl, const float* __restrict__ xr,
                                   const int* __restrict__ srcA, const int* __restrict__ dstA,
                                   const float* __restrict__ att, float* __restrict__ logits) {
  const int wave = blockIdx.x * (blockDim.x >> 5) + (threadIdx.x >> 5);
  const int lane = threadIdx.x & 31;
  if (wave >= ETOT) return;
  const float* pl = xl + (size_t)srcA[wave] * HC;
  const float* pr = xr + (size_t)dstA[wave] * HC;
  for (int h = 0; h < HEADS; ++h) {
    float p = 0.f;
    for (int j = 0; j < 4; ++j) {
      int c = lane + 32 * j;
      float u = pl[h * HID + c] + pr[h * HID + c];
      u = (u > 0.f) ? u : 0.2f * u;                // leaky_relu(0.2)
      p += u * att[h * HID + c];
    }
    p = wave_sum(p);
    if (lane == 0) logits[(size_t)wave * HEADS + h] = p;
  }
}

__global__ void edge_max_kernel(const float* __restrict__ logits, const int* __restrict__ dstA,
                                unsigned* __restrict__ menc) {
  int idx = blockIdx.x * blockDim.x + threadIdx.x;
  if (idx >= ETOT * HEADS) return;
  int e = idx >> 2, h = idx & 3;
  atomicMax(&menc[(size_t)dstA[e] * HEADS + h], fenc(logits[idx]));
}

__global__ void edge_exp_kernel(const float* __restrict__ logits, const int* __restrict__ dstA,
                                const unsigned* __restrict__ menc, float* __restrict__ exv,
                                float* __restrict__ den) {
  int idx = blockIdx.x * blockDim.x + threadIdx.x;
  if (idx >= ETOT * HEADS) return;
  int e = idx >> 2, h = idx & 3;
  int d = dstA[e];
  float m = fdec(menc[(size_t)d * HEADS + h]);
  float ex = expf(logits[idx] - m);
  exv[idx] = ex;
  atomicAdd(&den[(size_t)d * HEADS + h], ex);
}

__global__ void edge_scatter_kernel(const float* __restrict__ xl, const int* __restrict__ srcA,
                                    const int* __restrict__ dstA, const float* __restrict__ exv,
                                    const float* __restrict__ den, float* __restrict__ accum) {
  int e = blockIdx.x;
  int s = srcA[e], d = dstA[e];
  for (int c = threadIdx.x; c < HC; c += blockDim.x) {
    int h = c >> 7;
    float alpha = exv[(size_t)e * HEADS + h] / (den[(size_t)d * HEADS + h] + 1e-16f);
    atomicAdd(&accum[(size_t)d * HC + c], xl[(size_t)s * HC + c] * alpha);
  }
}

// head-mean + bo -> ELU -> +residual -> LayerNorm
__global__ void node_update(const float* __restrict__ accum, const float* __restrict__ bo,
                            const float* __restrict__ resid, const float* __restrict__ g,
                            const float* __restrict__ b, float* __restrict__ out, int nrows) {
  const int wave = blockIdx.x * (blockDim.x >> 5) + (threadIdx.x >> 5);
  const int lane = threadIdx.x & 31;
  if (wave >= nrows) return;
  const float* arow = accum + (size_t)wave * HC;
  float v[4], s = 0.f, s2 = 0.f;
  for (int j = 0; j < 4; ++j) {
    int c = lane + 32 * j;
    float t = 0.25f * (arow[c] + arow[HID + c] + arow[2 * HID + c] + arow[3 * HID + c]) + bo[c];
    t = elu1(t) + resid[(size_t)wave * HID + c];
    v[j] = t; s += t; s2 += t * t;
  }
  s = wave_sum(s); s2 = wave_sum(s2);
  float mean = s * (1.f / HID);
  float var  = s2 * (1.f / HID) - mean * mean;
  float inv  = rsqrtf(var + 1e-5f);
  for (int j = 0; j < 4; ++j) {
    int c = lane + 32 * j;
    out[(size_t)wave * HID + c] = (v[j] - mean) * inv * g[c] + b[c];
  }
}

// ---------------------------------------------------------------- MLP head
__global__ void head1_kernel(const float* __restrict__ h, const float* __restrict__ W,
                             const float* __restrict__ b, float* __restrict__ z) {
  int idx = blockIdx.x * blockDim.x + threadIdx.x;
  if (idx >= NRXN * 64) return;
  int r = idx >> 6, c = idx & 63;
  float acc = b[c];
  const float* hr = h + (size_t)r * HID;
  for (int k = 0; k < HID; ++k) acc += hr[k] * W[k * 64 + c];
  z[idx] = elu1(acc);
}

__global__ void head2_kernel(const float* __restrict__ z, const float* __restrict__ W,
                             const float* __restrict__ b, float* __restrict__ out) {
  int r = blockIdx.x * blockDim.x + threadIdx.x;
  if (r >= NRXN) return;
  float acc = b[0];
  const float* zr = z + (size_t)r * 64;
  for (int k = 0; k < 64; ++k) acc += zr[k] * W[k];
  out[r] = 1.f / (1.f + expf(-acc));
}

// ---------------------------------------------------------------- host side
static void run_layer(const float* xin, const float* Wl, const float* bl,
                      const float* Wr, const float* br, const float* att, const float* bo,
                      const float* g, const float* b, float* hout,
                      int* srcA, int* dstA, float* xl, float* xr, float* accum,
                      float* logits, float* exv, unsigned* menc, float* den,
                      hipStream_t stream) {
  dim3 B(256);
  const int tiles = (NNODE / 16) * (HC / 16);   // 60000 wave-tiles
  gemm_wmma_bias<<<(tiles + 7) / 8, B, 0, stream>>>(xin, Wl, bl, xl, NNODE, HID, HC);
  gemm_wmma_bias<<<(tiles + 7) / 8, B, 0, stream>>>(xin, Wr, br, xr, NNODE, HID, HC);
  edge_logits_kernel<<<(ETOT + 7) / 8, B, 0, stream>>>(xl, xr, srcA, dstA, att, logits);
  // menc & den are contiguous: zero both (encoded 0 == "below every real logit")
  zero_f32<<<((size_t)NNODE * HEADS * 2 + 255) / 256, B, 0, stream>>>((float*)menc, (size_t)NNODE * HEADS * 2);
  zero_f32<<<(((size_t)NNODE * HC) + 255) / 256, B, 0, stream>>>(accum, (size_t)NNODE * HC);
  edge_max_kernel<<<(ETOT * HEADS + 255) / 256, B, 0, stream>>>(logits, dstA, menc);
  edge_exp_kernel<<<(ETOT * HEADS + 255) / 256, B, 0, stream>>>(logits, dstA, menc, exv, den);
  edge_scatter_kernel<<<ETOT, B, 0, stream>>>(xl, srcA, dstA, exv, den, accum);
  node_update<<<(NNODE + 7) / 8, B, 0, stream>>>(accum, bo, xin, g, b, hout, NNODE);
}

extern "C" void kernel_launch(void* const* d_in, const int* in_sizes, int n_in,
                              void* d_out, int out_size, void* d_ws, size_t ws_size,
                              hipStream_t stream) {
  (void)in_sizes; (void)n_in; (void)out_size; (void)ws_size;
  const float* x_rxn  = (const float*)d_in[0];
  const float* x_met  = (const float*)d_in[1];
  const int* ei_rm_src = (const int*)d_in[2];
  const int* ei_rm_dst = (const int*)d_in[3];
  const int* ei_mr_src = (const int*)d_in[4];
  const int* ei_mr_dst = (const int*)d_in[5];
  const float* W_proj = (const float*)d_in[6];
  const float* b_proj = (const float*)d_in[7];
  const float* g_proj = (const float*)d_in[8];
  const float* be_proj = (const float*)d_in[9];
  const float* g1_Wl = (const float*)d_in[10];
  const float* g1_bl = (const float*)d_in[11];
  const float* g1_Wr = (const float*)d_in[12];
  const float* g1_br = (const float*)d_in[13];
  const float* g1_att = (const float*)d_in[14];
  const float* g1_bo = (const float*)d_in[15];
  const float* n1_g = (const float*)d_in[16];
  const float* n1_b = (const float*)d_in[17];
  const float* g2_Wl = (const float*)d_in[18];
  const float* g2_bl = (const float*)d_in[19];
  const float* g2_Wr = (const float*)d_in[20];
  const float* g2_br = (const float*)d_in[21];
  const float* g2_att = (const float*)d_in[22];
  const float* g2_bo = (const float*)d_in[23];
  const float* n2_g = (const float*)d_in[24];
  const float* n2_b = (const float*)d_in[25];
  const float* Wh1 = (const float*)d_in[26];
  const float* bh1 = (const float*)d_in[27];
  const float* Wh2 = (const float*)d_in[28];
  const float* bh2 = (const float*)d_in[29];
  float* out = (float*)d_out;

  // ---- workspace carve-out (floats) ----
  float* wsf = (float*)d_ws;
  size_t o = 0;
  int* srcA = (int*)(wsf + o); o += ETOT;
  int* dstA = (int*)(wsf + o); o += ETOT;
  float* xlin = wsf + o; o += (size_t)NNODE * HID;   // proj pre-LN; reused as h2
  float* xbuf = wsf + o; o += (size_t)NNODE * HID;   // x after proj LN+ELU
  float* h1   = wsf + o; o += (size_t)NNODE * HID;
  float* xl   = wsf + o; o += (size_t)NNODE * HC;
  float* xr   = wsf + o; o += (size_t)NNODE * HC;
  float* accum= wsf + o; o += (size_t)NNODE * HC;
  float* logits = wsf + o; o += (size_t)ETOT * HEADS;
  float* exv  = wsf + o; o += (size_t)ETOT * HEADS;
  unsigned* menc = (unsigned*)(wsf + o); o += (size_t)NNODE * HEADS;  // must precede den
  float* den  = wsf + o; o += (size_t)NNODE * HEADS;
  float* zbuf = wsf + o; o += (size_t)NRXN * 64;
  float* h2 = xlin;   // xlin dead after ln_elu_rows -> reuse for layer-2 output

  dim3 B(256);
  build_edges<<<(ETOT + 255) / 256, B, 0, stream>>>(ei_rm_src, ei_rm_dst, ei_mr_src, ei_mr_dst, srcA, dstA);
  proj_gemm_kernel<<<NNODE / 16, B, 0, stream>>>(x_rxn, x_met, W_proj, b_proj, xlin);
  ln_elu_rows<<<(NNODE + 7) / 8, B, 0, stream>>>(xlin, g_proj, be_proj, xbuf, NNODE);

  run_layer(xbuf, g1_Wl, g1_bl, g1_Wr, g1_br, g1_att, g1_bo, n1_g, n1_b, h1,
            srcA, dstA, xl, xr, accum, logits, exv, menc, den, stream);
  run_layer(h1, g2_Wl, g2_bl, g2_Wr, g2_br, g2_att, g2_bo, n2_g, n2_b, h2,
            srcA, dstA, xl, xr, accum, logits, exv, menc, den, stream);

  head1_kernel<<<(NRXN * 64 + 255) / 256, B, 0, stream>>>(h2, Wh1, bh1, zbuf);
  head2_kernel<<<(NRXN + 255) / 256, B, 0, stream>>>(zbuf, Wh2, bh2, out);
}